// RoiPoolLayer_73177652789762
// MI455X (gfx1250) — compile-verified
//
#include <hip/hip_runtime.h>
#include <float.h>

// RoIPool (Caffe-style, max) over two feature maps, summed.
// conv0: [2,256,50,68] stride-8 (scale 0.125)
// conv1: [2,256,25,34] stride-16 (scale 0.0625)
// rois : [128,5] = (batch_idx, x1, y1, x2, y2) in image coords
// out  : [128,256,7,7] f32

#define PH 7
#define PW 7
#define C_CH 256
#define H0 50
#define W0 68
#define H1 25
#define W1 34
#define N_ROIS 128
#define SCALE0 0.125f
#define SCALE1 0.0625f

// Classic RoIPool bin bounds, float32 math identical to the JAX reference:
// lo = clip(floor(p*bs)   + start, 0, limit)
// hi = clip(ceil((p+1)*bs)+ start, 0, limit)
__device__ __forceinline__ void bin_bounds(float start, float size, int p,
                                           float limit, int* lo, int* hi) {
    float bs  = size / 7.0f;
    float lof = floorf((float)p * bs) + start;
    float hif = ceilf((float)(p + 1) * bs) + start;
    lof = fminf(fmaxf(lof, 0.0f), limit);
    hif = fminf(fmaxf(hif, 0.0f), limit);
    *lo = (int)lof;
    *hi = (int)hif;
}

__global__ __launch_bounds__(256) void roipool2_sum_kernel(
    const float* __restrict__ conv0,
    const float* __restrict__ conv1,
    const float* __restrict__ rois,
    float* __restrict__ out)
{
    __shared__ int s_b0, s_b1;
    __shared__ int s_hlo0[PH], s_hhi0[PH], s_wlo0[PW], s_whi0[PW];
    __shared__ int s_hlo1[PH], s_hhi1[PH], s_wlo1[PW], s_whi1[PW];

    const int n   = blockIdx.x;   // roi index
    const int ph  = blockIdx.y;   // pooled row
    const int tid = threadIdx.x;

    // 7 threads compute all bin bounds for this ROI into LDS (both scales).
    // The ROI math is wave-uniform -> compiler scalarizes to s_mul_f32/s_rndne_f32.
    if (tid < 7) {
        const float* r = rois + n * 5;
        const float x1 = r[1], y1 = r[2], x2 = r[3], y2 = r[4];
        if (tid == 0) {
            const int b = (int)r[0];            // truncation == astype(int32)
            s_b0 = b * (C_CH * H0 * W0);
            s_b1 = b * (C_CH * H1 * W1);
        }
        // scale 0 (stride 8). rintf == round-half-to-even == jnp.round
        float sw = rintf(x1 * SCALE0), sh = rintf(y1 * SCALE0);
        float ew = rintf(x2 * SCALE0), eh = rintf(y2 * SCALE0);
        float rw = fmaxf(ew - sw + 1.0f, 1.0f);
        float rh = fmaxf(eh - sh + 1.0f, 1.0f);
        bin_bounds(sw, rw, tid, (float)W0, &s_wlo0[tid], &s_whi0[tid]);
        bin_bounds(sh, rh, tid, (float)H0, &s_hlo0[tid], &s_hhi0[tid]);
        // scale 1 (stride 16)
        sw = rintf(x1 * SCALE1); sh = rintf(y1 * SCALE1);
        ew = rintf(x2 * SCALE1); eh = rintf(y2 * SCALE1);
        rw = fmaxf(ew - sw + 1.0f, 1.0f);
        rh = fmaxf(eh - sh + 1.0f, 1.0f);
        bin_bounds(sw, rw, tid, (float)W1, &s_wlo1[tid], &s_whi1[tid]);
        bin_bounds(sh, rh, tid, (float)H1, &s_hlo1[tid], &s_hhi1[tid]);
    }
    __syncthreads();

    // h-bounds are block-uniform (fixed ph) -> scalar registers.
    const int hl0 = s_hlo0[ph], hh0 = s_hhi0[ph];
    const int hl1 = s_hlo1[ph], hh1 = s_hhi1[ph];
    const bool hrow0 = (hh0 > hl0);
    const bool hrow1 = (hh1 > hl1);
    const int b0 = s_b0, b1 = s_b1;

    // Flat work g = c*7 + pw, pw fastest -> lanes walk adjacent columns of the
    // same feature rows (line-local reads) and writes are near-contiguous.
    for (int g = tid; g < C_CH * PW; g += 256) {
        const int c  = g / 7;
        const int pw = g - c * 7;

        const int wl0 = s_wlo0[pw], wh0 = s_whi0[pw];
        const int wl1 = s_wlo1[pw], wh1 = s_whi1[pw];
        const bool has0 = hrow0 & (wh0 > wl0);
        const bool has1 = hrow1 & (wh1 > wl1);

        const float* base1 = conv1 + b1 + c * (H1 * W1) + hl1 * W1;
        if (has1) {
            // gfx1250 global_prefetch_b8: pull the stride-16 window toward the
            // WGP while the stride-8 max-loop below executes.
            __builtin_prefetch(base1 + wl1, 0, 0);
        }

        float acc = 0.0f;  // empty bins contribute exactly 0 (reference semantics)

        if (has0) {
            const float* row = conv0 + b0 + c * (H0 * W0) + hl0 * W0;
            float m = -FLT_MAX;  // == finfo(float32).min seed in reference
            for (int h = hl0; h < hh0; ++h, row += W0) {
                for (int w = wl0; w < wh0; ++w) m = fmaxf(m, row[w]);
            }
            acc += m;
        }
        if (has1) {
            const float* row = base1;
            float m = -FLT_MAX;
            for (int h = hl1; h < hh1; ++h, row += W1) {
                for (int w = wl1; w < wh1; ++w) m = fmaxf(m, row[w]);
            }
            acc += m;
        }

        // Streaming output, written once, never re-read by this kernel:
        // non-temporal store keeps the conv windows (reused across overlapping ROIs)
        // resident in cache instead of being rinsed by output traffic.
        __builtin_nontemporal_store(acc, &out[((n * C_CH + c) * PH + ph) * PW + pw]);
    }
}

extern "C" void kernel_launch(void* const* d_in, const int* in_sizes, int n_in,
                              void* d_out, int out_size, void* d_ws, size_t ws_size,
                              hipStream_t stream) {
    (void)in_sizes; (void)n_in; (void)out_size; (void)d_ws; (void)ws_size;
    const float* conv0 = (const float*)d_in[0];
    const float* conv1 = (const float*)d_in[1];
    const float* rois  = (const float*)d_in[2];
    float* out = (float*)d_out;

    dim3 grid(N_ROIS, PH, 1);   // 896 blocks
    dim3 block(256, 1, 1);      // 8 wave32s
    roipool2_sum_kernel<<<grid, block, 0, stream>>>(conv0, conv1, rois, out);
}